// ClusteringLayer_28810640622069
// MI455X (gfx1250) — compile-verified
//
#include <hip/hip_runtime.h>

typedef __attribute__((ext_vector_type(2))) float v2f;
typedef __attribute__((ext_vector_type(8))) float v8f;

#define DDIM 64
#define KC 100
#define KPAD 112            // 7 tiles of 16 columns
#define NTILES 7
#define LDS_STRIDE 68       // floats; multiple of 4 (16B-aligned f4 stores), bank-spreading
#define WAVES_PER_WG 4
#define WG_SIZE (WAVES_PER_WG * 32)

__global__ __launch_bounds__(WG_SIZE)
void ClusteringLayer_q_kernel(const float* __restrict__ X,
                              const float* __restrict__ C,
                              float* __restrict__ out,
                              int Nrows) {
  __shared__ float sC[KPAD * LDS_STRIDE];
  __shared__ float sCsq[KPAD];

  const int tid = threadIdx.x;

  // ---- Stage clusters into LDS, zero-padded to 112 rows (float4 vectors) ----
  {
    const float4* C4 = (const float4*)C;
    float4 z; z.x = z.y = z.z = z.w = 0.0f;
    for (int idx = tid; idx < KPAD * (DDIM / 4); idx += WG_SIZE) {
      int row = idx >> 4;
      int q4  = idx & 15;
      float4 v = (row < KC) ? C4[row * (DDIM / 4) + q4] : z;
      *(float4*)&sC[row * LDS_STRIDE + q4 * 4] = v;
    }
  }
  __syncthreads();
  if (tid < KPAD) {
    float s = 0.0f;
#pragma unroll 8
    for (int d = 0; d < DDIM; ++d) {
      float c = sC[tid * LDS_STRIDE + d];
      s += c * c;
    }
    // padded columns get a huge ||c||^2 so q ~ 0 and they drop out of the sum
    sCsq[tid] = (tid < KC) ? s : 3.0e38f;
  }
  __syncthreads();

  const int lane  = tid & 31;
  const int laneN = lane & 15;   // column-within-tile (B/C/D) or row-within-16 (A)
  const int hi    = lane >> 4;   // half-wave select
  const int wave  = tid >> 5;
  const long m0 = ((long)blockIdx.x * WAVES_PER_WG + wave) * 16;
  if (m0 >= Nrows) return;       // barriers already done above

  // ---- A fragments: 16 K-chunks of 4. f32 16x4 WMMA A layout:
  //      lanes 0-15: VGPR0=K0,VGPR1=K1 ; lanes 16-31: VGPR0=K2,VGPR1=K3
  const float* xrow = X + (m0 + laneN) * DDIM + 2 * hi;
  v2f a[16];
#pragma unroll
  for (int c = 0; c < 16; ++c)
    a[c] = *(const v2f*)(xrow + 4 * c);

  // ---- ||x||^2 per row from the A registers (each half-wave holds half the row)
  float part = 0.0f;
#pragma unroll
  for (int c = 0; c < 16; ++c) part += a[c].x * a[c].x + a[c].y * a[c].y;
  float xsq_full = part + __shfl_xor(part, 16, 32);  // lane L -> ||x_{m0+L%16}||^2
  float xs[8];
#pragma unroll
  for (int r = 0; r < 8; ++r)
    xs[r] = __shfl(xsq_full, r + 8 * hi, 32);        // broadcast into C/D row layout

  // ---- GEMM: cross = X * C^T over 7 column tiles, A reused from registers ----
  v8f acc[NTILES];
#pragma unroll
  for (int t = 0; t < NTILES; ++t) {
    v8f c0 = {};
    const float* bbase = &sC[(t * 16 + laneN) * LDS_STRIDE + 2 * hi];
#pragma unroll
    for (int c = 0; c < 16; ++c) {
      v2f b = *(const v2f*)(bbase + 4 * c);
      c0 = __builtin_amdgcn_wmma_f32_16x16x4_f32(
          /*neg_a=*/false, a[c], /*neg_b=*/false, b,
          /*c_mod=*/(short)0, c0, /*reuse_a=*/false, /*reuse_b=*/false);
    }
    acc[t] = c0;
  }

  // ---- q = 1/(1+d2), d2 = ||x||^2 + ||c||^2 - 2*cross  (ALPHA==1 -> power==1)
  float srow[8];
#pragma unroll
  for (int r = 0; r < 8; ++r) srow[r] = 0.0f;
#pragma unroll
  for (int t = 0; t < NTILES; ++t) {
    float csq = sCsq[t * 16 + laneN];
#pragma unroll
    for (int r = 0; r < 8; ++r) {
      float d2 = xs[r] + csq - 2.0f * acc[t][r];
      float q  = 1.0f / (1.0f + d2);
      acc[t][r] = q;
      srow[r] += q;
    }
  }
  // ---- horizontal row sums across the 16 lanes of each C/D group ----
#pragma unroll
  for (int r = 0; r < 8; ++r) {
    float s = srow[r];
    s += __shfl_xor(s, 1, 32);
    s += __shfl_xor(s, 2, 32);
    s += __shfl_xor(s, 4, 32);
    s += __shfl_xor(s, 8, 32);
    srow[r] = 1.0f / s;
  }
  // ---- normalize + coalesced stores (mask the partial last tile) ----
#pragma unroll
  for (int t = 0; t < NTILES; ++t) {
    int j = t * 16 + laneN;
    if (j < KC) {
#pragma unroll
      for (int r = 0; r < 8; ++r) {
        long m = m0 + r + 8 * hi;
        out[m * KC + j] = acc[t][r] * srow[r];
      }
    }
  }
}

extern "C" void kernel_launch(void* const* d_in, const int* in_sizes, int n_in,
                              void* d_out, int out_size, void* d_ws, size_t ws_size,
                              hipStream_t stream) {
  const float* X = (const float*)d_in[0];   // [N, 64] fp32
  const float* C = (const float*)d_in[1];   // [100, 64] fp32
  float* out = (float*)d_out;               // [N, 100] fp32

  int Nrows = in_sizes[0] / DDIM;
  int rowsPerWG = 16 * WAVES_PER_WG;
  int grid = (Nrows + rowsPerWG - 1) / rowsPerWG;

  hipLaunchKernelGGL(ClusteringLayer_q_kernel, dim3(grid), dim3(WG_SIZE), 0, stream,
                     X, C, out, Nrows);
}